// GCN_83270825935254
// MI455X (gfx1250) — compile-verified
//
#include <hip/hip_runtime.h>
#include <math.h>

#define NN 16384
#define EE 524288

typedef float v2f __attribute__((ext_vector_type(2)));
typedef float v8f __attribute__((ext_vector_type(8)));

// ---------------------------------------------------------------------------
// init: deg = 1.0 (self loop), xw1 = 0 (atomic target),
//       W1t[n][k] = (n<8) ? W1[k][n] : 0   -- padded transposed weights
// ---------------------------------------------------------------------------
__global__ __launch_bounds__(256) void k_init(float* __restrict__ deg,
                                              float* __restrict__ xw1,
                                              const float* __restrict__ W1,
                                              float* __restrict__ W1t) {
  int i = blockIdx.x * 256 + threadIdx.x;     // 0 .. 16*NN-1
  if (i < NN) deg[i] = 1.0f;
  if (i < NN * 8) xw1[i] = 0.0f;
  if (i < NN * 16) {
    int n = i >> 14;                          // NN == 1<<14
    int k = i & (NN - 1);
    W1t[i] = (n < 8) ? W1[(size_t)k * 8 + n] : 0.0f;
  }
}

// deg[dst] += 1 per edge
__global__ __launch_bounds__(256) void k_deg(const long long* __restrict__ dst,
                                             float* __restrict__ deg) {
  int e = blockIdx.x * 256 + threadIdx.x;
  if (e < EE) atomicAdd(&deg[(int)dst[e]], 1.0f);
}

// dinv = rsqrt(deg) in place
__global__ __launch_bounds__(256) void k_dinv(float* __restrict__ deg) {
  int i = blockIdx.x * 256 + threadIdx.x;
  if (i < NN) { float d = deg[i]; deg[i] = (d > 0.0f) ? rsqrtf(d) : 0.0f; }
}

// ---------------------------------------------------------------------------
// xw1 = x @ W1   via V_WMMA_F32_16X16X4_F32
//   8192 waves: 1024 row-tiles x 8 K-chunks of 2048; partials -> f32 atomics.
//   A fragment (16x4 f32): lane half hi picks K+2*hi, K+2*hi+1 -> one b64 load.
//   B fragment from padded transposed W1t[16][NN]: same addressing, rows 8..15
//   are zero so no masking / divergence is needed; EXEC stays all-ones.
// ---------------------------------------------------------------------------
__global__ __launch_bounds__(256) void k_gemm1(const float* __restrict__ x,
                                               const float* __restrict__ W1t,
                                               float* __restrict__ xw1) {
  const int lane = threadIdx.x & 31;
  const int wave = threadIdx.x >> 5;
  const int w    = blockIdx.x * 8 + wave;   // 0..8191
  const int tile = w & 1023;                // row tile
  const int kc   = w >> 10;                 // K chunk (0..7)
  const int r0   = tile * 16;
  const int kbeg = kc * 2048;

  const int hi = lane >> 4;                 // lane half: 0 or 1
  const int m  = lane & 15;

  const float* __restrict__ ap = x   + (size_t)(r0 + m) * NN + 2 * hi + kbeg;
  const float* __restrict__ bp = W1t + (size_t)m        * NN + 2 * hi + kbeg;

  v8f c = {0.f, 0.f, 0.f, 0.f, 0.f, 0.f, 0.f, 0.f};

  for (int k = 0; k < 2048; k += 16) {
    v2f a0 = *(const v2f*)(ap + k);
    v2f a1 = *(const v2f*)(ap + k + 4);
    v2f a2 = *(const v2f*)(ap + k + 8);
    v2f a3 = *(const v2f*)(ap + k + 12);
    v2f b0 = *(const v2f*)(bp + k);
    v2f b1 = *(const v2f*)(bp + k + 4);
    v2f b2 = *(const v2f*)(bp + k + 8);
    v2f b3 = *(const v2f*)(bp + k + 12);
    c = __builtin_amdgcn_wmma_f32_16x16x4_f32(false, a0, false, b0, (short)0, c, false, false);
    c = __builtin_amdgcn_wmma_f32_16x16x4_f32(false, a1, false, b1, (short)0, c, false, false);
    c = __builtin_amdgcn_wmma_f32_16x16x4_f32(false, a2, false, b2, (short)0, c, false, false);
    c = __builtin_amdgcn_wmma_f32_16x16x4_f32(false, a3, false, b3, (short)0, c, false, false);
  }

  // c[v] = D[r0 + v + 8*hi][m]; only columns 0..7 are real outputs
  if (m < 8) {
    #pragma unroll
    for (int v = 0; v < 8; ++v)
      atomicAdd(&xw1[(size_t)(r0 + v + 8 * hi) * 8 + m], c[v]);
  }
}

// acc1 = xw1 * dinv^2  (self-loop contribution)
__global__ __launch_bounds__(256) void k_self1(const float* __restrict__ xw1,
                                               const float* __restrict__ dinv,
                                               float* __restrict__ acc1) {
  int i = blockIdx.x * 256 + threadIdx.x;
  if (i < NN * 8) {
    float d = dinv[i >> 3];
    acc1[i] = xw1[i] * d * d;
  }
}

// acc1[dst] += xw1[src] * dinv[src]*dinv[dst]   (8 channels per edge)
__global__ __launch_bounds__(256) void k_edge1(const long long* __restrict__ src,
                                               const long long* __restrict__ dst,
                                               const float* __restrict__ dinv,
                                               const float* __restrict__ xw1,
                                               float* __restrict__ acc1) {
  int e = blockIdx.x * 256 + threadIdx.x;
  if (e >= EE) return;
  int s = (int)src[e], d = (int)dst[e];
  float nrm = dinv[s] * dinv[d];
  const float4* xs = (const float4*)(xw1 + (size_t)s * 8);
  float4 a = xs[0], b = xs[1];
  float* o = acc1 + (size_t)d * 8;
  atomicAdd(o + 0, a.x * nrm); atomicAdd(o + 1, a.y * nrm);
  atomicAdd(o + 2, a.z * nrm); atomicAdd(o + 3, a.w * nrm);
  atomicAdd(o + 4, b.x * nrm); atomicAdd(o + 5, b.y * nrm);
  atomicAdd(o + 6, b.z * nrm); atomicAdd(o + 7, b.w * nrm);
}

// h = tanh(acc1 + b1), in place
__global__ __launch_bounds__(256) void k_act1(float* __restrict__ acc1,
                                              const float* __restrict__ b1) {
  int i = blockIdx.x * 256 + threadIdx.x;
  if (i < NN * 8) acc1[i] = tanhf(acc1[i] + b1[i & 7]);
}

// layer 2: xw2 = h @ W2 (K=8, VALU); acc2 = xw2 * dinv^2
__global__ __launch_bounds__(256) void k_layer2(const float* __restrict__ h,
                                                const float* __restrict__ W2,
                                                const float* __restrict__ dinv,
                                                float* __restrict__ xw2,
                                                float* __restrict__ acc2) {
  int i = blockIdx.x * 256 + threadIdx.x;
  if (i >= NN) return;
  const float4* hp = (const float4*)(h + (size_t)i * 8);
  float4 a = hp[0], b = hp[1];
  float z0 = a.x * W2[0] + a.y * W2[2] + a.z * W2[4] + a.w * W2[6]
           + b.x * W2[8] + b.y * W2[10] + b.z * W2[12] + b.w * W2[14];
  float z1 = a.x * W2[1] + a.y * W2[3] + a.z * W2[5] + a.w * W2[7]
           + b.x * W2[9] + b.y * W2[11] + b.z * W2[13] + b.w * W2[15];
  float d2 = dinv[i] * dinv[i];
  xw2[i * 2 + 0] = z0;       xw2[i * 2 + 1] = z1;
  acc2[i * 2 + 0] = z0 * d2; acc2[i * 2 + 1] = z1 * d2;
}

// acc2[dst] += xw2[src] * nrm  (2 channels per edge)
__global__ __launch_bounds__(256) void k_edge2(const long long* __restrict__ src,
                                               const long long* __restrict__ dst,
                                               const float* __restrict__ dinv,
                                               const float* __restrict__ xw2,
                                               float* __restrict__ acc2) {
  int e = blockIdx.x * 256 + threadIdx.x;
  if (e >= EE) return;
  int s = (int)src[e], d = (int)dst[e];
  float nrm = dinv[s] * dinv[d];
  float2 v = *(const float2*)(xw2 + (size_t)s * 2);
  atomicAdd(&acc2[(size_t)d * 2 + 0], v.x * nrm);
  atomicAdd(&acc2[(size_t)d * 2 + 1], v.y * nrm);
}

// emb = tanh(acc2 + b2); out = sigmoid(emb @ Wc + bc)
__global__ __launch_bounds__(256) void k_head(const float* __restrict__ acc2,
                                              const float* __restrict__ b2,
                                              const float* __restrict__ Wc,
                                              const float* __restrict__ bc,
                                              float* __restrict__ out) {
  int i = blockIdx.x * 256 + threadIdx.x;
  if (i >= NN) return;
  float e0 = tanhf(acc2[i * 2 + 0] + b2[0]);
  float e1 = tanhf(acc2[i * 2 + 1] + b2[1]);
  float z  = e0 * Wc[0] + e1 * Wc[1] + bc[0];
  out[i] = 1.0f / (1.0f + expf(-z));
}

// ---------------------------------------------------------------------------
extern "C" void kernel_launch(void* const* d_in, const int* in_sizes, int n_in,
                              void* d_out, int out_size, void* d_ws, size_t ws_size,
                              hipStream_t stream) {
  const float*     x  = (const float*)d_in[0];
  const long long* ei = (const long long*)d_in[1];   // int64 [2,E]
  const float*     W1 = (const float*)d_in[2];
  const float*     b1 = (const float*)d_in[3];
  const float*     W2 = (const float*)d_in[4];
  const float*     b2 = (const float*)d_in[5];
  const float*     Wc = (const float*)d_in[6];
  const float*     bc = (const float*)d_in[7];
  float*           out = (float*)d_out;

  const long long* src = ei;
  const long long* dst = ei + EE;

  float* ws   = (float*)d_ws;
  float* dinv = ws;                 // [N]        deg, then rsqrt in place
  float* W1t  = dinv + NN;          // [16, N]    padded transposed W1
  float* xw1  = W1t + NN * 16;      // [N, 8]
  float* acc1 = xw1 + NN * 8;       // [N, 8]     becomes h
  float* xw2  = acc1 + NN * 8;      // [N, 2]
  float* acc2 = xw2 + NN * 2;       // [N, 2]

  const int T = 256;
  k_init  <<<(NN * 16 + T - 1) / T, T, 0, stream>>>(dinv, xw1, W1, W1t);
  k_deg   <<<(EE + T - 1) / T,      T, 0, stream>>>(dst, dinv);
  k_dinv  <<<(NN + T - 1) / T,      T, 0, stream>>>(dinv);
  k_gemm1 <<<1024,                  T, 0, stream>>>(x, W1t, xw1);  // 8192 waves
  k_self1 <<<(NN * 8 + T - 1) / T,  T, 0, stream>>>(xw1, dinv, acc1);
  k_edge1 <<<(EE + T - 1) / T,      T, 0, stream>>>(src, dst, dinv, xw1, acc1);
  k_act1  <<<(NN * 8 + T - 1) / T,  T, 0, stream>>>(acc1, b1);
  k_layer2<<<(NN + T - 1) / T,      T, 0, stream>>>(acc1, W2, dinv, xw2, acc2);
  k_edge2 <<<(EE + T - 1) / T,      T, 0, stream>>>(src, dst, dinv, xw2, acc2);
  k_head  <<<(NN + T - 1) / T,      T, 0, stream>>>(acc2, b2, Wc, bc, out);
}